// SpatialGCN_50723563766080
// MI455X (gfx1250) — compile-verified
//
#include <hip/hip_runtime.h>

// ---------------------------------------------------------------------------
// SpatialGCN for MI455X (gfx1250), 3-pass scheme:
//  P1: qkv projection via v_wmma_f32_16x16x4_f32, store k, accumulate
//      A=q.v^T, Gk=k.k^T, ksum with in-register WMMA accumulators + atomics.
//  P2: one small block: softmax + analytic BatchNorm stats from Gram matrices,
//      folding the whole tail into out = relu(F_b @ k + d2 + x).
//  P3: [128x64]x[64xN] WMMA GEMM + residual + relu.
// ---------------------------------------------------------------------------

typedef __attribute__((ext_vector_type(2))) float v2f;
typedef __attribute__((ext_vector_type(8))) float v8f;

#define P_DIM 128
#define C_DIM 64
#define N_DIM 110592   // 48*48*48
#define B_DIM 2
#define TILE_N 64
#define LDQ 72         // padded LDS row stride for the [192 x 64] qkv tile
#define EPSV 1e-5f

// workspace layout (float offsets)
#define OFF_KBUF  0
#define KBUF_SZ   (B_DIM * C_DIM * N_DIM)            // 14,155,776 floats
#define OFF_AQV   (OFF_KBUF + KBUF_SZ)               // [2][64][64]
#define OFF_GK    (OFF_AQV + B_DIM * C_DIM * C_DIM)  // [2][64][64]
#define OFF_KSUM  (OFF_GK  + B_DIM * C_DIM * C_DIM)  // [2][64]
#define OFF_S     (OFF_KSUM + B_DIM * C_DIM)
#define OFF_M1    (OFF_S   + B_DIM * C_DIM * C_DIM)
#define OFF_M2    (OFF_M1  + B_DIM * C_DIM * C_DIM)  // [2][128][64]
#define OFF_T     (OFF_M2  + B_DIM * P_DIM * C_DIM)  // scratch [2][128][64]
#define OFF_F     (OFF_T   + B_DIM * P_DIM * C_DIM)  // [2][128][64]
#define OFF_D2    (OFF_F   + B_DIM * P_DIM * C_DIM)  // [128]
#define OFF_MU1   (OFF_D2  + P_DIM)
#define OFF_SC1   (OFF_MU1 + C_DIM)
#define OFF_SH1   (OFF_SC1 + C_DIM)
#define OFF_C2    (OFF_SH1 + C_DIM)
#define OFF_MV    (OFF_C2  + P_DIM)
#define OFF_SC2   (OFF_MV  + P_DIM)

static __device__ __forceinline__ v8f wmma_f32(v2f a, v2f b, v8f c) {
  // 8 args per probe: (neg_a, A, neg_b, B, c_mod, C, reuse_a, reuse_b)
  return __builtin_amdgcn_wmma_f32_16x16x4_f32(false, a, false, b, (short)0, c,
                                               false, false);
}

// A- or B-fragment where the contraction (K) index runs along a row (stride 1).
// ISA layout: lanes 0-15 hold K={k0,k0+1}, lanes 16-31 hold K={k0+2,k0+3};
// the row index (M for A, N-col for B) is lane&15.
static __device__ __forceinline__ v2f frag_k_contig(const float* t, int ld,
                                                    int rbase, int k0, int lane) {
  const int r  = rbase + (lane & 15);
  const int kc = k0 + ((lane >> 4) << 1);
  const float* p = t + (size_t)r * ld + kc;
  v2f f; f.x = p[0]; f.y = p[1];
  return f;
}

// B-fragment where K runs down the column (row stride ldn), columns stride 1.
static __device__ __forceinline__ v2f frag_k_strided(const float* t, int ldn,
                                                     int k0, int cbase, int lane) {
  const int kc = k0 + ((lane >> 4) << 1);
  const int c  = cbase + (lane & 15);
  const float* p = t + (size_t)kc * ldn + c;
  v2f f; f.x = p[0]; f.y = p[ldn];
  return f;
}

// ------------------------- Pass 0: zero reduction buffers -------------------
__global__ void zero_kernel(float* __restrict__ ws) {
  const int n = 2 * B_DIM * C_DIM * C_DIM + B_DIM * C_DIM;  // Aqv+Gk+ksum
  for (int i = threadIdx.x + blockIdx.x * blockDim.x; i < n;
       i += blockDim.x * gridDim.x)
    ws[OFF_AQV + i] = 0.0f;
}

// ------------------------- Pass 1: projections + Grams ----------------------
__global__ __launch_bounds__(512) void proj_gram_kernel(
    const float* __restrict__ x,
    const float* __restrict__ Wk, const float* __restrict__ bk,
    const float* __restrict__ Wq, const float* __restrict__ bq,
    const float* __restrict__ Wv, const float* __restrict__ bv,
    float* __restrict__ ws, int tiles_per_group) {
  __shared__ float qkv[192 * LDQ];  // rows 0-63 = k, 64-127 = q, 128-191 = v
  const int tid  = threadIdx.x;
  const int lane = tid & 31;
  const int wave = tid >> 5;        // 16 waves
  const int bidx = blockIdx.y;
  const float* xb = x + (size_t)bidx * P_DIM * N_DIM;
  float* kbuf = ws + OFF_KBUF + (size_t)bidx * C_DIM * N_DIM;
  float* Aqv  = ws + OFF_AQV + bidx * C_DIM * C_DIM;
  float* Gk   = ws + OFF_GK  + bidx * C_DIM * C_DIM;
  float* ksum = ws + OFF_KSUM + bidx * C_DIM;

  v8f accA = {};  // wave's 16x16 block of A = q.v^T
  v8f accG = {};  // wave's 16x16 block of Gk = k.k^T
  float ks = 0.0f;
  const int gi = (wave >> 2) & 3;   // gram block row
  const int gj = wave & 3;          // gram block col

  for (int it = 0; it < tiles_per_group; ++it) {
    const int tile  = blockIdx.x * tiles_per_group + it;
    const int nbase = tile * TILE_N;

    // phase 1: 48 (16-row x 16-col) projection subtiles; 3 per wave
    for (int ss = 0; ss < 3; ++ss) {
      const int s  = wave + ss * 16;   // 0..47
      const int rt = s >> 2;           // 0..11 (row tile of 192)
      const int ct = s & 3;            // 0..3  (col tile of 64)
      const float* W; const float* bias; int brow;
      if (rt < 4)      { W = Wk; bias = bk; brow = rt * 16; }
      else if (rt < 8) { W = Wq; bias = bq; brow = (rt - 4) * 16; }
      else             { W = Wv; bias = bv; brow = (rt - 8) * 16; }
      v8f acc = {};
      for (int k0 = 0; k0 < P_DIM; k0 += 4) {
        v2f a = frag_k_contig(W, P_DIM, brow, k0, lane);
        v2f b = frag_k_strided(xb, N_DIM, k0, nbase + ct * 16, lane);
        acc = wmma_f32(a, b, acc);
      }
      const int halo = (lane >> 4) << 3;  // upper lanes hold M+8
      for (int r = 0; r < 8; ++r) {
        const int rl = r + halo;
        qkv[(rt * 16 + rl) * LDQ + ct * 16 + (lane & 15)] =
            acc[r] + bias[brow + rl];
      }
    }
    __syncthreads();

    // phase 2a: spill k rows (0..63) to HBM (only big intermediate)
    for (int idx = tid; idx < C_DIM * TILE_N; idx += 512) {
      const int c = idx >> 6, j = idx & 63;
      kbuf[(size_t)c * N_DIM + nbase + j] = qkv[c * LDQ + j];
    }
    // phase 2b: per-channel running sum of k
    if (tid < C_DIM) {
      float s = 0.0f;
      for (int j = 0; j < TILE_N; ++j) s += qkv[tid * LDQ + j];
      ks += s;
    }
    // phase 2c: Gram contractions over the 64 columns (K = n)
    for (int kk = 0; kk < TILE_N; kk += 4) {
      v2f aq = frag_k_contig(qkv, LDQ, 64  + gi * 16, kk, lane);   // q rows
      v2f bw = frag_k_contig(qkv, LDQ, 128 + gj * 16, kk, lane);   // v rows (B)
      accA = wmma_f32(aq, bw, accA);
      v2f ak = frag_k_contig(qkv, LDQ, gi * 16, kk, lane);         // k rows
      v2f bk2 = frag_k_contig(qkv, LDQ, gj * 16, kk, lane);        // k rows (B)
      accG = wmma_f32(ak, bk2, accG);
    }
    __syncthreads();
  }

  // flush: one atomic per accumulator element
  const int halo = (lane >> 4) << 3;
  const int col  = gj * 16 + (lane & 15);
  for (int r = 0; r < 8; ++r) {
    const int row = gi * 16 + r + halo;
    atomicAdd(&Aqv[row * C_DIM + col], accA[r]);
    atomicAdd(&Gk[row * C_DIM + col], accG[r]);
  }
  if (tid < C_DIM) atomicAdd(&ksum[tid], ks);
}

// --------- Pass 2: softmax + analytic BN folding (single workgroup) ---------
__global__ __launch_bounds__(256) void stats_kernel(
    const float* __restrict__ Wg, const float* __restrict__ g_wg,
    const float* __restrict__ b_wg, const float* __restrict__ Wo,
    const float* __restrict__ bo, const float* __restrict__ g_out,
    const float* __restrict__ b_out, float* __restrict__ ws) {
  const int tid = threadIdx.x;
  float* Aqv = ws + OFF_AQV;  float* Gk  = ws + OFF_GK;
  float* ksum = ws + OFF_KSUM; float* S  = ws + OFF_S;
  float* M1 = ws + OFF_M1;    float* M2  = ws + OFF_M2;
  float* T  = ws + OFF_T;     float* F   = ws + OFF_F;
  float* d2 = ws + OFF_D2;    float* mu1 = ws + OFF_MU1;
  float* sc1 = ws + OFF_SC1;  float* sh1 = ws + OFF_SH1;
  float* c2 = ws + OFF_C2;    float* mv  = ws + OFF_MV;
  float* sc2 = ws + OFF_SC2;
  const float invBN = 1.0f / ((float)B_DIM * (float)N_DIM);

  // softmax over last dim of A[b][c][:]
  if (tid < B_DIM * C_DIM) {
    const float* row = Aqv + tid * C_DIM;
    float* srow = S + tid * C_DIM;
    float m = row[0];
    for (int j = 1; j < C_DIM; ++j) m = fmaxf(m, row[j]);
    float sum = 0.0f;
    for (int j = 0; j < C_DIM; ++j) { float e = __expf(row[j] - m); srow[j] = e; sum += e; }
    const float inv = 1.0f / sum;
    for (int j = 0; j < C_DIM; ++j) srow[j] *= inv;
  }
  __syncthreads();
  // M1[b][d][c] = sum_e Wg[d][e] * S[b][c][e]   (AVW = M1 @ k)
  for (int idx = tid; idx < B_DIM * C_DIM * C_DIM; idx += 256) {
    const int b = idx >> 12, rem = idx & 4095, d = rem >> 6, c = rem & 63;
    const float* wr = Wg + d * C_DIM;
    const float* sr = S + b * 4096 + c * C_DIM;
    float s = 0.0f;
    for (int e = 0; e < C_DIM; ++e) s += wr[e] * sr[e];
    M1[idx] = s;
  }
  __syncthreads();
  // T = M1_b @ Gk_b
  for (int idx = tid; idx < B_DIM * C_DIM * C_DIM; idx += 256) {
    const int b = idx >> 12, rem = idx & 4095, d = rem >> 6, c = rem & 63;
    const float* mr = M1 + b * 4096 + d * C_DIM;
    const float* gc = Gk + b * 4096 + c;
    float s = 0.0f;
    for (int e = 0; e < C_DIM; ++e) s += mr[e] * gc[e * C_DIM];
    T[idx] = s;
  }
  __syncthreads();
  if (tid < C_DIM) {  // BN1 stats: mean = M1@ksum/(bN); E[x^2] = diag(M1 Gk M1^T)/(bN)
    float m = 0.0f, e2 = 0.0f;
    for (int b = 0; b < B_DIM; ++b) {
      const float* mr = M1 + b * 4096 + tid * C_DIM;
      const float* tr = T  + b * 4096 + tid * C_DIM;
      const float* kr = ksum + b * C_DIM;
      for (int c = 0; c < C_DIM; ++c) { m += mr[c] * kr[c]; e2 += tr[c] * mr[c]; }
    }
    m *= invBN; e2 *= invBN;
    mu1[tid] = m;
    const float sc = g_wg[tid] * rsqrtf((e2 - m * m) + EPSV);
    sc1[tid] = sc;
    sh1[tid] = b_wg[tid] - m * sc;
  }
  __syncthreads();
  // M2[b][p][c] = sum_d Wo[p][d]*sc1[d]*M1[b][d][c] ;  c2 = Wo@sh1 + bo
  for (int idx = tid; idx < B_DIM * P_DIM * C_DIM; idx += 256) {
    const int b = idx >> 13, rem = idx & 8191, p = rem >> 6, c = rem & 63;
    const float* wr = Wo + p * C_DIM;
    const float* m1b = M1 + b * 4096;
    float s = 0.0f;
    for (int d = 0; d < C_DIM; ++d) s += wr[d] * sc1[d] * m1b[d * C_DIM + c];
    M2[idx] = s;
  }
  if (tid < P_DIM) {
    const float* wr = Wo + tid * C_DIM;
    float s = 0.0f;
    for (int d = 0; d < C_DIM; ++d) s += wr[d] * sh1[d];
    c2[tid] = s + bo[tid];
  }
  __syncthreads();
  // T = M2_b @ Gk_b
  for (int idx = tid; idx < B_DIM * P_DIM * C_DIM; idx += 256) {
    const int b = idx >> 13, rem = idx & 8191, p = rem >> 6, c = rem & 63;
    const float* mr = M2 + b * 8192 + p * C_DIM;
    const float* gc = Gk + b * 4096 + c;
    float s = 0.0f;
    for (int e = 0; e < C_DIM; ++e) s += mr[e] * gc[e * C_DIM];
    T[idx] = s;
  }
  __syncthreads();
  if (tid < P_DIM) {  // BN2 stats: var = diag(M2 Gk M2^T)/(bN) - mv^2 (shift c2 cancels)
    float m = 0.0f, e2 = 0.0f;
    for (int b = 0; b < B_DIM; ++b) {
      const float* mr = M2 + b * 8192 + tid * C_DIM;
      const float* tr = T  + b * 8192 + tid * C_DIM;
      const float* kr = ksum + b * C_DIM;
      for (int c = 0; c < C_DIM; ++c) { m += mr[c] * kr[c]; e2 += tr[c] * mr[c]; }
    }
    m *= invBN; e2 *= invBN;
    mv[tid] = m;
    const float sc = g_out[tid] * rsqrtf((e2 - m * m) + EPSV);
    sc2[tid] = sc;
    d2[tid] = b_out[tid] - sc * m;  // = b_out + sc*(c2 - mu2), mu2 = mv + c2
  }
  __syncthreads();
  for (int idx = tid; idx < B_DIM * P_DIM * C_DIM; idx += 256) {
    const int p = (idx & 8191) >> 6;
    F[idx] = sc2[p] * M2[idx];
  }
}

// -------------- Pass 3: out = relu(F_b @ k + d2 + x), WMMA GEMM -------------
__global__ __launch_bounds__(512) void out_kernel(const float* __restrict__ x,
                                                  const float* __restrict__ ws,
                                                  float* __restrict__ out) {
  const int tid = threadIdx.x, lane = tid & 31, wave = tid >> 5;
  const int bidx  = blockIdx.y;
  const int nbase = blockIdx.x * TILE_N;
  const float* kbuf = ws + OFF_KBUF + (size_t)bidx * C_DIM * N_DIM;
  const float* F  = ws + OFF_F + bidx * P_DIM * C_DIM;
  const float* d2 = ws + OFF_D2;
  const float* xb = x + (size_t)bidx * P_DIM * N_DIM;
  float* ob = out + (size_t)bidx * P_DIM * N_DIM;

  for (int ss = 0; ss < 2; ++ss) {
    const int s = wave + ss * 16;  // 32 subtiles: 8 row-tiles x 4 col-tiles
    const int rt = s >> 2, ct = s & 3;
    v8f acc = {};
    for (int k0 = 0; k0 < C_DIM; k0 += 4) {
      v2f a = frag_k_contig(F, C_DIM, rt * 16, k0, lane);
      v2f b = frag_k_strided(kbuf, N_DIM, k0, nbase + ct * 16, lane);
      acc = wmma_f32(a, b, acc);
    }
    const int halo = (lane >> 4) << 3;
    const int col  = nbase + ct * 16 + (lane & 15);
    for (int r = 0; r < 8; ++r) {
      const int p = rt * 16 + r + halo;
      const float v = acc[r] + d2[p] + xb[(size_t)p * N_DIM + col];
      ob[(size_t)p * N_DIM + col] = fmaxf(v, 0.0f);
    }
  }
}

extern "C" void kernel_launch(void* const* d_in, const int* in_sizes, int n_in,
                              void* d_out, int out_size, void* d_ws,
                              size_t ws_size, hipStream_t stream) {
  const float* x    = (const float*)d_in[0];
  const float* Wk   = (const float*)d_in[1];
  const float* bk   = (const float*)d_in[2];
  const float* Wq   = (const float*)d_in[3];
  const float* bq   = (const float*)d_in[4];
  const float* Wv   = (const float*)d_in[5];
  const float* bv   = (const float*)d_in[6];
  const float* Wg   = (const float*)d_in[7];
  const float* g_wg = (const float*)d_in[8];
  const float* b_wg = (const float*)d_in[9];
  const float* Wo   = (const float*)d_in[10];
  const float* bo   = (const float*)d_in[11];
  const float* g_out = (const float*)d_in[12];
  const float* b_out = (const float*)d_in[13];
  float* ws  = (float*)d_ws;
  float* out = (float*)d_out;

  zero_kernel<<<32, 256, 0, stream>>>(ws);

  const int tiles = N_DIM / TILE_N;  // 1728
  const int tiles_per_group = 8;
  dim3 g1(tiles / tiles_per_group, B_DIM);  // (216, 2)
  proj_gram_kernel<<<g1, 512, 0, stream>>>(x, Wk, bk, Wq, bq, Wv, bv, ws,
                                           tiles_per_group);
  stats_kernel<<<1, 256, 0, stream>>>(Wg, g_wg, b_wg, Wo, bo, g_out, b_out, ws);
  dim3 g3(tiles, B_DIM);
  out_kernel<<<g3, 512, 0, stream>>>(x, ws, out);
}